// TensorProductContractionLayer_71966472012432
// MI455X (gfx1250) — compile-verified
//
#include <hip/hip_runtime.h>

// Triple contraction  res[b,m,d,f] = sum_{a,c} x[b,m,a,f] y[b,m,c,f] cgc[a,d,c]
// as per-(b,m,ftile) GEMM  D[16x16] = W[16x256] * Z[256x16]  with
// W[d, a*16+c] = cgc[a,d,c]  and  Z[(a,c), f] = x[a,f]*y[c,f] formed in VGPRs.
// Uses V_WMMA_F32_16X16X4_F32 (f32 in/out, matches reference precision).

typedef __attribute__((ext_vector_type(2))) float v2f;
typedef __attribute__((ext_vector_type(8))) float v8f;

#define B_ 8
#define M_ 512
#define A_ 16
#define F_ 128

// LDS layout for W: [a][d][c] with d-stride 17 floats -> 17*L mod 64 covers all
// 16 distinct banks for lanes 0..15 (conflict-free ds loads).
#define WSTRIDE 17
#define WLDS_PER_A (A_ * WSTRIDE)      // 272 floats per a-slice
#define WLDS_TOTAL (A_ * WLDS_PER_A)   // 4352 floats = 17 KB

__global__ __launch_bounds__(256)
void tpcl_wmma_f32_kernel(const float* __restrict__ x,
                          const float* __restrict__ y,
                          const float* __restrict__ cgc,
                          float* __restrict__ out)
{
    __shared__ float w[WLDS_TOTAL];

    // Cooperative fill of cgc (4096 floats) into bank-padded LDS, once per block.
    for (int i = threadIdx.x; i < A_ * A_ * A_; i += 256) {
        int a = i >> 8;
        int d = (i >> 4) & 15;
        int c = i & 15;
        w[a * WLDS_PER_A + d * WSTRIDE + c] = cgc[i];
    }
    __syncthreads();

    const int lane      = threadIdx.x & 31;
    const int waveInBlk = threadIdx.x >> 5;
    const int gw        = blockIdx.x * 8 + waveInBlk;   // 32768 waves total
    const int bm        = gw >> 3;                      // 0..4095  (b*M+m)
    const int ftile     = gw & 7;                       // 0..7
    const int fcol      = (ftile << 4) + (lane & 15);   // this lane's f column
    const int hi        = lane >> 4;                    // half-wave select
    const int off       = hi << 1;                      // K-offset 0 or 2
    const int dM        = lane & 15;                    // this lane's d row (A op)

    const float* __restrict__ xp = x + (size_t)bm * (A_ * F_) + fcol;
    const float* __restrict__ yp = y + (size_t)bm * (A_ * F_) + fcol;

    // Full x column (16 values, one per a); coalesced 64B per half-wave.
    float xr[16];
#pragma unroll
    for (int a = 0; a < 16; ++a) xr[a] = xp[a * F_];

    // Each lane only ever needs c in {4k+off, 4k+off+1}: 8 y values.
    float yA[4], yB[4];
#pragma unroll
    for (int k = 0; k < 4; ++k) {
        yA[k] = yp[(4 * k + off)     * F_];
        yB[k] = yp[(4 * k + off + 1) * F_];
    }

    v8f acc = {};
    const float* __restrict__ wbase = &w[dM * WSTRIDE + off];

    // 64 K-chunks of 4 over k=(a,c): chunk t -> a = t>>2, c0 = 4*(t&3).
#pragma unroll
    for (int t = 0; t < 64; ++t) {
        const int a  = t >> 2;      // compile-time after unroll
        const int k4 = t & 3;       // compile-time after unroll

        // A operand: lane L holds W[d=L, c0+off], W[d=L, c0+off+1]
        v2f aop;
        aop.x = wbase[a * WLDS_PER_A + 4 * k4];
        aop.y = wbase[a * WLDS_PER_A + 4 * k4 + 1];

        // B operand: lane holds Z[k, f] = x[a,f]*y[c,f] for its two K rows
        v2f bop;
        bop.x = xr[a] * yA[k4];
        bop.y = xr[a] * yB[k4];

        acc = __builtin_amdgcn_wmma_f32_16x16x4_f32(
            /*neg_a=*/false, aop, /*neg_b=*/false, bop,
            /*c_mod=*/(short)0, acc, /*reuse_a=*/false, /*reuse_b=*/false);
    }

    // D layout: VGPR i -> (d = i + 8*hi, f = fcol); 64B-coalesced per half-wave.
    float* __restrict__ op = out + (size_t)bm * (A_ * F_) + (hi ? 8 * F_ : 0) + fcol;
#pragma unroll
    for (int i = 0; i < 8; ++i) op[i * F_] = acc[i];
}

extern "C" void kernel_launch(void* const* d_in, const int* in_sizes, int n_in,
                              void* d_out, int out_size, void* d_ws, size_t ws_size,
                              hipStream_t stream) {
    const float* x   = (const float*)d_in[0];
    const float* y   = (const float*)d_in[1];
    const float* cgc = (const float*)d_in[2];
    float* out = (float*)d_out;

    (void)in_sizes; (void)n_in; (void)out_size; (void)d_ws; (void)ws_size;

    // 4096 blocks x 256 threads = 32768 waves = (B*M) x 8 f-tiles
    dim3 grid(B_ * M_);
    dim3 block(256);
    tpcl_wmma_f32_kernel<<<grid, block, 0, stream>>>(x, y, cgc, out);
}